// GetDensity_32452772888585
// MI455X (gfx1250) — compile-verified
//
#include <hip/hip_runtime.h>

#define NTYPE    8
#define NWAVE    8
#define NIPSIN   3
#define NORBIT   128
#define N_OC     2
#define NLOCAL   10000
#define NALL     10000
#define MAXNEIGH 500000
#define NANG     13            // 1 + 3 + 9 angular rows
#define ROWSZ    (NANG*NWAVE)  // 104 floats per atom
#define PI_F     3.14159265358979323846f

typedef float v2f __attribute__((ext_vector_type(2)));
typedef float v8f __attribute__((ext_vector_type(8)));

// ---------------------------------------------------------------------------
// init: orb_coeff = params[species], zero sw and swsum accumulators
// ---------------------------------------------------------------------------
__global__ void k_init(const float* __restrict__ params,
                       const int*   __restrict__ species,
                       float* __restrict__ coeff,
                       float* __restrict__ sw,
                       float* __restrict__ swsum) {
    int i = blockIdx.x * blockDim.x + threadIdx.x;
    if (i < NLOCAL * ROWSZ) { sw[i] = 0.0f; swsum[i] = 0.0f; }
    if (i < NLOCAL * NWAVE) {
        int a = i / NWAVE, k = i % NWAVE;
        coeff[i] = params[species[a] * NWAVE + k];
    }
}

// ---------------------------------------------------------------------------
// edge pass: rebuild per-edge orbital (angular x radial [+ swsum feedback]),
// scale by orb_coeff[neigh], scatter-add into sw[center] (L2-resident atomics)
// ---------------------------------------------------------------------------
__global__ void k_edge(const float* __restrict__ cart,
                       const float* __restrict__ rs,
                       const float* __restrict__ inta,
                       const int*   __restrict__ atom_index,
                       const int*   __restrict__ species,
                       const int*   __restrict__ neigh_list,
                       const float* __restrict__ coeff,
                       const float* __restrict__ swsum,
                       float* __restrict__ sw,
                       int use_fb) {
    int e = blockIdx.x * blockDim.x + threadIdx.x;
    if (e >= MAXNEIGH) return;

    int c0 = atom_index[e];              // center (segment id)
    int c1 = atom_index[MAXNEIGH + e];
    int j  = neigh_list[e];              // neighbor local atom
    int s  = species[j];

    float dx = cart[c0*3+0] - cart[c1*3+0];
    float dy = cart[c0*3+1] - cart[c1*3+1];
    float dz = cart[c0*3+2] - cart[c1*3+2];
    float d  = sqrtf(dx*dx + dy*dy + dz*dz);
    float fc = 0.5f * __cosf(d * (PI_F / 5.0f)) + 0.5f;
    float dcut = fc * fc;

    float rad[NWAVE], cf[NWAVE];
#pragma unroll
    for (int k = 0; k < NWAVE; ++k) {
        float t = d - rs[s*NWAVE + k];
        rad[k]  = __expf(inta[s*NWAVE + k] * t * t);
        cf[k]   = coeff[j*NWAVE + k];
    }

    float dv[3] = {dx, dy, dz};
    float ang[NANG];
    ang[0] = dcut;
#pragma unroll
    for (int i = 0; i < 3; ++i) ang[1+i] = dcut * dv[i];
#pragma unroll
    for (int i = 0; i < 3; ++i)
#pragma unroll
        for (int q = 0; q < 3; ++q) ang[4 + i*3 + q] = ang[1+i] * dv[q];

    float*       dst = sw    + (size_t)c0 * ROWSZ;
    const float* fb  = swsum + (size_t)j  * ROWSZ;
    if (use_fb) __builtin_prefetch(fb, 0, 3);   // global_prefetch_b8

#pragma unroll
    for (int r = 0; r < NANG; ++r) {
#pragma unroll
        for (int k = 0; k < NWAVE; ++k) {
            float o = ang[r] * rad[k];
            if (use_fb) o += fb[r*NWAVE + k] * dcut;
            atomicAdd(dst + r*NWAVE + k, o * cf[k]);
        }
    }
}

// ---------------------------------------------------------------------------
// atom pass: density[i,m] = sum_r ( sum_k sw[i,r,k] * hyper[p(r),k,m] )^2
// via V_WMMA_F32_16X16X4_F32 (K=8 as two chained K=4 WMMAs).
// One wave handles a 16-atom tile; 8 N-tiles of 16 over NORBIT=128.
// A layout (ISA 7.12.2, 32-bit A 16x4): lanes0-15 M, v0=K0 v1=K1;
//                                       lanes16-31     v0=K2 v1=K3.
// B layout (4x16): lanes0-15 v0=rowK0 v1=rowK1; lanes16-31 v0=K2 v1=K3.
// C/D layout: lane L, vgpr v -> M = v + 8*(L>=16), N = L&15.
// ---------------------------------------------------------------------------
__global__ void __launch_bounds__(256)
k_atom_wmma(const float* __restrict__ sw,
            const float* __restrict__ hyper_t,   // [NIPSIN][NWAVE][NORBIT]
            float* __restrict__ dens_out) {
    int lane = threadIdx.x & 31;
    int wave = threadIdx.x >> 5;
    int tile = blockIdx.x * 8 + wave;
    if (tile >= NLOCAL / 16) return;            // wave-uniform exit; EXEC stays full

    int atom0 = tile * 16;
    int n  = lane & 15;                          // M for A-frag, N for B/C frags
    int hi = lane >> 4;                          // lane half
    int kb = hi * 2;                             // K sub-offset for this half

    // Preload all 13 A fragments (K split into [0..3] and [4..7])
    v2f a_lo[NANG], a_hi[NANG];
    const float* abase = sw + (size_t)(atom0 + n) * ROWSZ;
#pragma unroll
    for (int r = 0; r < NANG; ++r) {
        a_lo[r] = *(const v2f*)(abase + r*NWAVE + kb);
        a_hi[r] = *(const v2f*)(abase + r*NWAVE + 4 + kb);
    }

#pragma unroll
    for (int nt = 0; nt < 8; ++nt) {
        v8f dens = {0.f,0.f,0.f,0.f,0.f,0.f,0.f,0.f};
#pragma unroll
        for (int r = 0; r < NANG; ++r) {
            int p = (r == 0) ? 0 : ((r < 4) ? 1 : 2);   // INDEX_PARA
            const float* H = hyper_t + (size_t)p * NWAVE * NORBIT + nt*16 + n;
            v2f b_lo, b_hi;
            b_lo.x = H[(0 + kb) * NORBIT];
            b_lo.y = H[(1 + kb) * NORBIT];
            b_hi.x = H[(4 + kb) * NORBIT];
            b_hi.y = H[(5 + kb) * NORBIT];
            v8f c = {0.f,0.f,0.f,0.f,0.f,0.f,0.f,0.f};
            c = __builtin_amdgcn_wmma_f32_16x16x4_f32(
                    false, a_lo[r], false, b_lo, (short)0, c, false, false);
            c = __builtin_amdgcn_wmma_f32_16x16x4_f32(
                    false, a_hi[r], false, b_hi, (short)0, c, false, false);
            dens += c * c;                       // sum of squares over r
        }
#pragma unroll
        for (int v = 0; v < 8; ++v)
            dens_out[(size_t)(atom0 + v + 8*hi) * NORBIT + nt*16 + n] = dens[v];
    }
}

// ---------------------------------------------------------------------------
// coeff update: orb_coeff[i,k] += sum_m density[i,m]*W[m,k] + b[k]
// ---------------------------------------------------------------------------
__global__ void k_coeff(const float* __restrict__ dens,
                        const float* __restrict__ W,     // [NORBIT][NWAVE]
                        const float* __restrict__ b,     // [NWAVE]
                        float* __restrict__ coeff) {
    int i = blockIdx.x * blockDim.x + threadIdx.x;
    if (i >= NLOCAL * NWAVE) return;
    int a = i / NWAVE, k = i % NWAVE;
    const float* dr = dens + (size_t)a * NORBIT;
    float acc = b[k];
#pragma unroll 8
    for (int m = 0; m < NORBIT; ++m) acc += dr[m] * W[m*NWAVE + k];
    coeff[i] += acc;
}

// ---------------------------------------------------------------------------
// swsum += sw; sw = 0   (prepare feedback + clean accumulator for next round)
// ---------------------------------------------------------------------------
__global__ void k_accum(float* __restrict__ swsum, float* __restrict__ sw) {
    int i = blockIdx.x * blockDim.x + threadIdx.x;
    if (i < NLOCAL * ROWSZ) { swsum[i] += sw[i]; sw[i] = 0.0f; }
}

// ---------------------------------------------------------------------------
extern "C" void kernel_launch(void* const* d_in, const int* in_sizes, int n_in,
                              void* d_out, int out_size, void* d_ws, size_t ws_size,
                              hipStream_t stream) {
    const float* cart       = (const float*)d_in[0];
    const float* rs         = (const float*)d_in[1];
    const float* inta       = (const float*)d_in[2];
    const float* params     = (const float*)d_in[3];
    const float* hyper      = (const float*)d_in[4];   // [3][3][8][128]
    const float* W_oc       = (const float*)d_in[5];   // [2][128][8]
    const float* b_oc       = (const float*)d_in[6];   // [2][8]
    const int*   atom_index = (const int*)d_in[7];     // [2][MAXNEIGH]
    const int*   species    = (const int*)d_in[8];
    const int*   neigh_list = (const int*)d_in[9];
    float*       out        = (float*)d_out;           // [NLOCAL][NORBIT]

    // workspace layout (floats), all L2-resident working sets
    float* ws    = (float*)d_ws;
    float* coeff = ws;                                  //  80,000
    float* sw    = ws + 131072;                         // 1,040,000
    float* swsum = ws + 131072 + 1048576;               // 1,040,000
    float* dens  = ws + 131072 + 2*1048576;             // 1,280,000

    const int TB = 256;
    k_init<<<(NLOCAL*ROWSZ + TB-1)/TB, TB, 0, stream>>>(params, species, coeff, sw, swsum);

    for (int t = 0; t < N_OC + 1; ++t) {
        k_edge<<<(MAXNEIGH + TB-1)/TB, TB, 0, stream>>>(
            cart, rs, inta, atom_index, species, neigh_list,
            coeff, swsum, sw, (t > 0) ? 1 : 0);

        const float* hyper_t = hyper + (size_t)t * NIPSIN * NWAVE * NORBIT;
        float* dout = (t == N_OC) ? out : dens;
        int ntiles = NLOCAL / 16;                       // 625
        k_atom_wmma<<<(ntiles + 7)/8, TB, 0, stream>>>(sw, hyper_t, dout);

        if (t < N_OC) {
            k_coeff<<<(NLOCAL*NWAVE + TB-1)/TB, TB, 0, stream>>>(
                dens, W_oc + (size_t)t*NORBIT*NWAVE, b_oc + (size_t)t*NWAVE, coeff);
            k_accum<<<(NLOCAL*ROWSZ + TB-1)/TB, TB, 0, stream>>>(swsum, sw);
        }
    }
}